// DeformableConv2DLayer_4544075399270
// MI455X (gfx1250) — compile-verified
//
#include <hip/hip_runtime.h>
#include <hip/hip_bf16.h>

typedef __attribute__((ext_vector_type(2))) float v2f;
typedef __attribute__((ext_vector_type(8))) float v8f;

#define N_   4
#define CIN  256
#define H_   64
#define W_   64
#define COUT 256
#define KW_  3
#define KTAPS 9
#define CK   (CIN * KTAPS)   // 2304
#define P_   (H_ * W_)       // 4096

// ---------------------------------------------------------------------------
// Kernel 1: modulated deformable im2col (bilinear sampling, zero padding).
// Writes pair-interleaved layout colPair[ck>>1][p][ck&1] so that the GEMM's
// B operand (two consecutive K values at one pixel) is a single b64 load.
// ---------------------------------------------------------------------------
__global__ __launch_bounds__(256) void dcn_im2col_kernel(
    const float* __restrict__ inputs,   // [N][CIN][H][W]
    const float* __restrict__ offset,   // [N][2*KTAPS][H][W]
    const float* __restrict__ mask,     // [N][KTAPS][H][W]
    float* __restrict__ colp,           // [CK/2][P][2] (batch n scratch)
    int n)
{
    int p  = (blockIdx.x & 15) * 256 + threadIdx.x;  // pixel 0..4095
    int ck = blockIdx.x >> 4;                        // 0..2303
    int c  = ck / KTAPS;
    int k  = ck - c * KTAPS;
    int py = p >> 6, px = p & 63;
    int ki = k / KW_, kj = k - ki * KW_;

    const float* offb = offset + (size_t)n * (2 * KTAPS) * H_ * W_;
    float dy = offb[((2 * k    ) * H_ + py) * W_ + px];
    float dx = offb[((2 * k + 1) * H_ + py) * W_ + px];
    float mv = mask[(((size_t)n * KTAPS + k) * H_ + py) * W_ + px];

    float y = (float)(py - 1 + ki) + dy;   // stride=1, pad=1, dil=1
    float x = (float)(px - 1 + kj) + dx;
    float y0f = floorf(y), x0f = floorf(x);
    int   y0  = (int)y0f,  x0  = (int)x0f;
    float ly  = y - y0f,   lx  = x - x0f;

    const float* img = inputs + ((size_t)n * CIN + c) * (H_ * W_);
    bool yin0 = (y0     >= 0) & (y0     < H_);
    bool yin1 = (y0 + 1 >= 0) & (y0 + 1 < H_);
    bool xin0 = (x0     >= 0) & (x0     < W_);
    bool xin1 = (x0 + 1 >= 0) & (x0 + 1 < W_);
    float v00 = (yin0 & xin0) ? img[y0 * W_ + x0]           : 0.f;
    float v01 = (yin0 & xin1) ? img[y0 * W_ + x0 + 1]       : 0.f;
    float v10 = (yin1 & xin0) ? img[(y0 + 1) * W_ + x0]     : 0.f;
    float v11 = (yin1 & xin1) ? img[(y0 + 1) * W_ + x0 + 1] : 0.f;

    float val = ((v00 * (1.f - ly) + v10 * ly) * (1.f - lx) +
                 (v01 * (1.f - ly) + v11 * ly) * lx) * mv;

    colp[((size_t)(ck >> 1)) * (P_ * 2) + (size_t)p * 2 + (ck & 1)] = val;
}

// ---------------------------------------------------------------------------
// Kernel 2: Out[256][4096] = Filt[256][2304] x Col[2304][4096]
// via V_WMMA_F32_16X16X4_F32. One wave computes a 32x64 output tile:
// 2 A fragments (rows m..m+15, m+16..m+31) share 4 B fragments per K-step
// => 8 WMMAs per 6 b64 loads (0.75 loads/WMMA).
//
// Operand layouts (CDNA5 ISA, f32 16x16x4):
//   A: lanes 0-15 hold M=lane, K={k,k+1}; lanes 16-31 hold K={k+2,k+3}.
//   B: lanes 0-15 hold N=lane, K={k,k+1}; lanes 16-31 hold K={k+2,k+3}.
//   C/D: VGPR v, lanes 0-15 -> M=v, N=lane; lanes 16-31 -> M=v+8.
// ---------------------------------------------------------------------------
__global__ __launch_bounds__(256) void dcn_gemm_f32_wmma_kernel(
    const float* __restrict__ filt,   // [COUT][CK] (natural [Cout][Cin][3][3])
    const float* __restrict__ colp,   // [CK/2][P][2]
    float* __restrict__ out,          // [COUT][P] (batch n slice)
    int /*n*/)
{
    int wave  = blockIdx.x * (blockDim.x >> 5) + (threadIdx.x >> 5);
    int mTile = wave >> 6;         // 0..7   (Cout/32)
    int nTile = wave & 63;         // 0..63  (P/64)
    int lane  = threadIdx.x & 31;
    int half  = lane >> 4;         // which K pair this lane carries
    int lm    = lane & 15;

    int mBase = mTile * 32;
    int pBase = nTile * 64;

    // Per-lane A pointers (two 16-row fragments). kk even => 8B-aligned b64.
    const float* Ap0 = filt + (size_t)(mBase      + lm) * CK + half * 2;
    const float* Ap1 = filt + (size_t)(mBase + 16 + lm) * CK + half * 2;
    // Per-lane B pointer: pair-row offset 'half', pixel (pBase+lm).
    const float* Bp  = colp + (size_t)half * (P_ * 2) + (size_t)(pBase + lm) * 2;

    v8f acc0[4] = {{}, {}, {}, {}};
    v8f acc1[4] = {{}, {}, {}, {}};

    for (int k = 0; k < CK; k += 4) {
        v2f a0 = *(const v2f*)(Ap0 + k);
        v2f a1 = *(const v2f*)(Ap1 + k);
        const float* Brow = Bp + (size_t)(k >> 1) * (P_ * 2);
        v2f b0 = *(const v2f*)(Brow);
        v2f b1 = *(const v2f*)(Brow + 32);   // +16 pixels * 2 floats
        v2f b2 = *(const v2f*)(Brow + 64);
        v2f b3 = *(const v2f*)(Brow + 96);
        acc0[0] = __builtin_amdgcn_wmma_f32_16x16x4_f32(false, a0, false, b0, (short)0, acc0[0], false, false);
        acc0[1] = __builtin_amdgcn_wmma_f32_16x16x4_f32(false, a0, false, b1, (short)0, acc0[1], false, false);
        acc0[2] = __builtin_amdgcn_wmma_f32_16x16x4_f32(false, a0, false, b2, (short)0, acc0[2], false, false);
        acc0[3] = __builtin_amdgcn_wmma_f32_16x16x4_f32(false, a0, false, b3, (short)0, acc0[3], false, false);
        acc1[0] = __builtin_amdgcn_wmma_f32_16x16x4_f32(false, a1, false, b0, (short)0, acc1[0], false, false);
        acc1[1] = __builtin_amdgcn_wmma_f32_16x16x4_f32(false, a1, false, b1, (short)0, acc1[1], false, false);
        acc1[2] = __builtin_amdgcn_wmma_f32_16x16x4_f32(false, a1, false, b2, (short)0, acc1[2], false, false);
        acc1[3] = __builtin_amdgcn_wmma_f32_16x16x4_f32(false, a1, false, b3, (short)0, acc1[3], false, false);
    }

    // Store D: VGPR v -> row base + v + half*8, column pBase + t*16 + lm.
    float* orow0 = out + (size_t)(mBase      + half * 8) * P_ + pBase + lm;
    float* orow1 = out + (size_t)(mBase + 16 + half * 8) * P_ + pBase + lm;
#pragma unroll
    for (int v = 0; v < 8; ++v) {
#pragma unroll
        for (int t = 0; t < 4; ++t) {
            orow0[(size_t)v * P_ + t * 16] = acc0[t][v];
            orow1[(size_t)v * P_ + t * 16] = acc1[t][v];
        }
    }
}

// ---------------------------------------------------------------------------
// Host-side launcher. Per batch: im2col -> gemm, stream-ordered, reusing the
// 37.7 MB col scratch in d_ws (fits entirely in the 192 MB L2, so the GEMM's
// B reads never leave the cache).
// ---------------------------------------------------------------------------
extern "C" void kernel_launch(void* const* d_in, const int* in_sizes, int n_in,
                              void* d_out, int out_size, void* d_ws, size_t ws_size,
                              hipStream_t stream) {
    (void)in_sizes; (void)n_in; (void)out_size; (void)ws_size;
    const float* inputs = (const float*)d_in[0];
    const float* filt   = (const float*)d_in[1];
    const float* offset = (const float*)d_in[2];
    const float* mask   = (const float*)d_in[3];
    float*       out    = (float*)d_out;
    float*       colp   = (float*)d_ws;   // CK * P_ floats = 37,748,736 B

    for (int n = 0; n < N_; ++n) {
        dcn_im2col_kernel<<<dim3(CK * (P_ / 256)), 256, 0, stream>>>(
            inputs, offset, mask, colp, n);
        // (Cout/32)=8 M-tiles * 64 N-tiles = 512 waves; 8 waves per block.
        dcn_gemm_f32_wmma_kernel<<<dim3(512 / 8), 256, 0, stream>>>(
            filt, colp, out + (size_t)n * COUT * P_, n);
    }
}